// AttentionSortNet_1580547971899
// MI455X (gfx1250) — compile-verified
//
#include <hip/hip_runtime.h>
#include <hip/hip_bf16.h>

#define HEADS 8
#define BUCKETS 64
#define DIM 64
#define SEQ 8192
#define BH 32
#define ROWS_PER_BUCKET 128   // SEQ / BUCKETS
#define TEMPERATURE 0.7f
#define SINKHORN_ITER 8
#define EPS 1e-06f

typedef float v2f __attribute__((ext_vector_type(2)));
typedef float v8f __attribute__((ext_vector_type(8)));

// ---------------------------------------------------------------------------
// Phase 1: bucket means + positional embedding.
// grid = (BUCKETS, BH, 2), block = 256.
// Each block streams 128 rows x 64 dims (32 KB) of q or k with float4 loads,
// reduces over rows, scales by 1/128, adds pos emb, writes one 64-float row.
// ---------------------------------------------------------------------------
__global__ void __launch_bounds__(256)
bucket_mean_kernel(const float* __restrict__ q, const float* __restrict__ k,
                   const float* __restrict__ qpos, const float* __restrict__ kpos,
                   float* __restrict__ sq, float* __restrict__ sk) {
    const int bucket = blockIdx.x;          // 0..63
    const int bh     = blockIdx.y;          // 0..31
    const int which  = blockIdx.z;          // 0 = q, 1 = k
    const int tid    = threadIdx.x;         // 0..255

    const float* __restrict__ src = which ? k : q;
    const float* __restrict__ pos = which ? kpos : qpos;
    float* __restrict__ dst       = which ? sk : sq;

    // float4 view of this bucket's 128x64 slab: 2048 float4s, 8 per thread.
    const float4* __restrict__ base =
        (const float4*)(src + ((size_t)bh * SEQ + (size_t)bucket * ROWS_PER_BUCKET) * DIM);

    float ax = 0.f, ay = 0.f, az = 0.f, aw = 0.f;
#pragma unroll
    for (int i = 0; i < 8; ++i) {
        float4 v = base[tid + i * 256];     // stride 256 keeps (idx & 15) == dim-group
        ax += v.x; ay += v.y; az += v.z; aw += v.w;
    }

    __shared__ float4 red[256];
    red[tid] = make_float4(ax, ay, az, aw);
    __syncthreads();

    // Tree-reduce the 16 partials per dim-group (offsets are multiples of 16,
    // so tid & 15 — the dim-group — is preserved).
#pragma unroll
    for (int off = 128; off >= 16; off >>= 1) {
        if (tid < off) {
            float4 a = red[tid], b = red[tid + off];
            red[tid] = make_float4(a.x + b.x, a.y + b.y, a.z + b.z, a.w + b.w);
        }
        __syncthreads();
    }

    if (tid < 16) {
        const float inv = 1.0f / (float)ROWS_PER_BUCKET;
        const int h = bh & (HEADS - 1);
        float4 m = red[tid];
        const float4 p = ((const float4*)pos)[((size_t)h * BUCKETS + bucket) * (DIM / 4) + tid];
        float4 o = make_float4(m.x * inv + p.x, m.y * inv + p.y,
                               m.z * inv + p.z, m.w * inv + p.w);
        ((float4*)dst)[((size_t)bh * BUCKETS + bucket) * (DIM / 4) + tid] = o;
    }
}

// ---------------------------------------------------------------------------
// Phase 2: R = sq . sk^T, batched 64x64x64 fp32 GEMM via V_WMMA_F32_16X16X4_F32.
// grid = (16 tiles, 32 batches), block = 32 (one wave per 16x16 output tile,
// EXEC all-ones as required by WMMA).
// A layout (16x4 f32): lane<16 -> (M=lane,  K=kk..kk+1), lane>=16 -> (M=lane-16, K=kk+2..kk+3)
// B layout (4x16 f32): lane<16 -> (N=lane,  K=kk..kk+1), lane>=16 -> (N=lane-16, K=kk+2..kk+3)
// C layout: VGPR v -> M = v + 8*(lane>=16), N = lane&15.
// ---------------------------------------------------------------------------
__global__ void __launch_bounds__(32)
gemm_wmma_kernel(const float* __restrict__ sq, const float* __restrict__ sk,
                 float* __restrict__ R) {
    const int tile = blockIdx.x;            // 0..15
    const int b    = blockIdx.y;            // 0..31
    const int tm   = tile >> 2;
    const int tn   = tile & 3;
    const int lane = threadIdx.x;           // 0..31
    const int half = lane >> 4;             // 0 or 1
    const int l    = lane & 15;

    const float* __restrict__ A = sq + (size_t)b * BUCKETS * DIM;   // row-major (bucket, dim)
    const float* __restrict__ B = sk + (size_t)b * BUCKETS * DIM;

    const float* arow = A + (size_t)(tm * 16 + l) * DIM + half * 2; // A[M=row][K]
    const float* brow = B + (size_t)(tn * 16 + l) * DIM + half * 2; // B[K][N] = sk[N][K]

    v8f c = {};
#pragma unroll
    for (int kk = 0; kk < DIM; kk += 4) {
        v2f a  = *(const v2f*)(arow + kk);
        v2f bb = *(const v2f*)(brow + kk);
        c = __builtin_amdgcn_wmma_f32_16x16x4_f32(
                /*neg_a=*/false, a, /*neg_b=*/false, bb,
                /*c_mod=*/(short)0, c, /*reuse_a=*/false, /*reuse_b=*/false);
    }

    float* out = R + (size_t)b * (BUCKETS * BUCKETS)
                   + (size_t)(tm * 16 + half * 8) * BUCKETS + tn * 16 + l;
#pragma unroll
    for (int v = 0; v < 8; ++v) out[(size_t)v * BUCKETS] = c[v];
}

// ---------------------------------------------------------------------------
// Phase 3: gumbel-sinkhorn. grid = (32 batches), block = 64.
// r = (log(relu(R)+eps)+gumbel)/T in LDS (pitch 65 -> bank-conflict-free rows),
// 8 iterations of row-LSE then col-LSE subtraction, output exp(r).
// ---------------------------------------------------------------------------
__global__ void __launch_bounds__(64)
sinkhorn_kernel(const float* __restrict__ R, const float* __restrict__ gumbel,
                float* __restrict__ out) {
    const int b = blockIdx.x;               // 0..31
    const int t = threadIdx.x;              // 0..63
    __shared__ float r[BUCKETS * (BUCKETS + 1)];

    const float* __restrict__ Rb = R + (size_t)b * BUCKETS * BUCKETS;
    const float* __restrict__ Gb = gumbel + (size_t)b * BUCKETS * BUCKETS;

    const float invT = 1.0f / TEMPERATURE;
#pragma unroll 4
    for (int i = 0; i < BUCKETS; ++i) {     // coalesced: thread t -> column t
        float v = Rb[i * BUCKETS + t];
        v = __logf(fmaxf(v, 0.0f) + EPS);
        r[i * (BUCKETS + 1) + t] = (v + Gb[i * BUCKETS + t]) * invT;
    }
    __syncthreads();

    for (int it = 0; it < SINKHORN_ITER; ++it) {
        // axis=2 (rows): thread t owns row t
        {
            float m = -INFINITY;
#pragma unroll 4
            for (int j = 0; j < BUCKETS; ++j) m = fmaxf(m, r[t * (BUCKETS + 1) + j]);
            float s = 0.0f;
#pragma unroll 4
            for (int j = 0; j < BUCKETS; ++j) s += __expf(r[t * (BUCKETS + 1) + j] - m);
            const float lse = m + __logf(s);
#pragma unroll 4
            for (int j = 0; j < BUCKETS; ++j) r[t * (BUCKETS + 1) + j] -= lse;
        }
        __syncthreads();
        // axis=1 (cols): thread t owns column t
        {
            float m = -INFINITY;
#pragma unroll 4
            for (int i = 0; i < BUCKETS; ++i) m = fmaxf(m, r[i * (BUCKETS + 1) + t]);
            float s = 0.0f;
#pragma unroll 4
            for (int i = 0; i < BUCKETS; ++i) s += __expf(r[i * (BUCKETS + 1) + t] - m);
            const float lse = m + __logf(s);
#pragma unroll 4
            for (int i = 0; i < BUCKETS; ++i) r[i * (BUCKETS + 1) + t] -= lse;
        }
        __syncthreads();
    }

    float* __restrict__ Ob = out + (size_t)b * BUCKETS * BUCKETS;
#pragma unroll 4
    for (int i = 0; i < BUCKETS; ++i)
        Ob[i * BUCKETS + t] = __expf(r[i * (BUCKETS + 1) + t]);
}

// ---------------------------------------------------------------------------
extern "C" void kernel_launch(void* const* d_in, const int* in_sizes, int n_in,
                              void* d_out, int out_size, void* d_ws, size_t ws_size,
                              hipStream_t stream) {
    const float* q      = (const float*)d_in[0];   // (32, 8192, 64)
    const float* k      = (const float*)d_in[1];   // (32, 8192, 64)
    const float* qpos   = (const float*)d_in[2];   // (1, 8, 64, 64)
    const float* kpos   = (const float*)d_in[3];   // (1, 8, 64, 64)
    const float* gumbel = (const float*)d_in[4];   // (32, 64, 64)
    float* out          = (float*)d_out;           // (32, 64, 64)

    const size_t smat_elems = (size_t)BH * BUCKETS * DIM;      // 131072 floats
    float* sq = (float*)d_ws;
    float* sk = sq + smat_elems;
    float* R  = sk + smat_elems;                               // 1.5 MB total

    // Phase 1: bucket means + pos emb (bandwidth-bound streaming pass)
    bucket_mean_kernel<<<dim3(BUCKETS, BH, 2), 256, 0, stream>>>(q, k, qpos, kpos, sq, sk);

    // Phase 2: batched 64x64x64 fp32 GEMM on the matrix cores
    gemm_wmma_kernel<<<dim3(16, BH), 32, 0, stream>>>(sq, sk, R);

    // Phase 3: gumbel-sinkhorn + exp
    sinkhorn_kernel<<<dim3(BH), 64, 0, stream>>>(R, gumbel, out);
}